// VectorQuantizer_3504693313641
// MI455X (gfx1250) — compile-verified
//
#include <hip/hip_runtime.h>

// ---------------------------------------------------------------------------
// VQ-VAE vector quantizer, fused for MI455X (gfx1250, wave32).
//
// Roofline: ~670MB of traffic (distances 268MB + encodings 268MB + quantized
// 67MB + inputs 68MB) -> ~29us floor at 23.3 TB/s. GEMM is only 34.4 GFLOP,
// so everything is fused into one streaming pass.
// Precision: codes are ~1e-3 apart -> fp32 WMMA (v_wmma_f32_16x16x4_f32).
// Argmin: each lane owns columns {n : n%16==laneLo}; running per-row min is
// kept in registers across all 64 tiles; 16-lane shuffle reduction runs once.
// ---------------------------------------------------------------------------

#define K_CODES   1024
#define DIMS      256
#define N_ROWS    65536        // 64*32*32
#define M_TILE    128
#define N_BLK     64
#define ROWSTRIDE 260          // 256+4: 16B-multiple rows (b128 fill) AND
                               // conflict-free 16-lane b64 fragment loads
#define BLOCK     256          // 8 waves

typedef __attribute__((ext_vector_type(2))) float v2f;
typedef __attribute__((ext_vector_type(4))) float v4f;
typedef __attribute__((ext_vector_type(8))) float v8f;

// d_ws layout (floats): [0,1024) wn ; [1024,2048) counts(u32) ; [2048,2560) loss partials

__global__ __launch_bounds__(256) void vq_prep(const float* __restrict__ w,
                                               float* __restrict__ wsbuf) {
  int k = blockIdx.x * blockDim.x + threadIdx.x;
  if (k < K_CODES) {
    const float* row = w + (long long)k * DIMS;
    float s = 0.f;
#pragma unroll 4
    for (int d = 0; d < DIMS; ++d) s = fmaf(row[d], row[d], s);
    wsbuf[k] = s;
    ((unsigned int*)wsbuf)[K_CODES + k] = 0u;   // zero histogram each launch
  }
}

__global__ __launch_bounds__(BLOCK) void vq_main(const float* __restrict__ x,
                                                 const float* __restrict__ w,
                                                 float* __restrict__ out,
                                                 float* __restrict__ wsbuf) {
  extern __shared__ float lds[];
  float* xs  = lds;                         // M_TILE * ROWSTRIDE
  float* wsh = xs + M_TILE * ROWSTRIDE;     // N_BLK * ROWSTRIDE
  float* xn  = wsh + N_BLK * ROWSTRIDE;     // M_TILE
  float* wn  = xn + M_TILE;                 // K_CODES
  int*   rowIdx = (int*)(wn + K_CODES);     // M_TILE
  float* red = (float*)(rowIdx + M_TILE);   // 8

  const int t      = threadIdx.x;
  const int lane   = t & 31;
  const int wave   = t >> 5;
  const int laneLo = lane & 15;
  const int hi     = lane >> 4;

  const long long rowBase = (long long)blockIdx.x * M_TILE;
  const int b   = (int)(rowBase >> 10);
  const int hw0 = (int)(rowBase & 1023);

  // output slices (flat tuple concat)
  float* dist   = out;                                    // [N,K]
  float* quant  = out + 67108864LL;                       // [64,256,32,32]
  float* enc    = out + 83886081LL;                       // [N,K]
  float* encIdx = out + 150994945LL;                      // [N]

  // ---- load x tile (NCHW -> [row][d] in LDS), coalesced on hw ----
  const float* xin = x + (long long)b * DIMS * 1024 + hw0;
  for (int it = 0; it < (M_TILE * DIMS) / BLOCK; ++it) {
    int e = it * BLOCK + t;
    int d = e >> 7, r = e & 127;
    xs[r * ROWSTRIDE + d] = xin[(long long)d * 1024 + r];
  }
  for (int i = t; i < K_CODES; i += BLOCK) wn[i] = wsbuf[i];
  __syncthreads();

  // ---- per-row ||x||^2 ----
  if (t < M_TILE) {
    const float* row = xs + t * ROWSTRIDE;
    float s = 0.f;
#pragma unroll 4
    for (int d = 0; d < DIMS; ++d) s = fmaf(row[d], row[d], s);
    xn[t] = s;
  }
  __syncthreads();

  const int mr = wave * 16;                 // each wave owns one 16-row tile
  float xnr[8];
#pragma unroll
  for (int i = 0; i < 8; ++i) xnr[i] = xn[mr + i + 8 * hi];

  float bestv[8]; int besti[8];
#pragma unroll
  for (int i = 0; i < 8; ++i) { bestv[i] = 3.4e38f; besti[i] = 0; }

  // ---- GEMM + distances + per-lane running argmin ----
  for (int nb = 0; nb < K_CODES / N_BLK; ++nb) {
    const int n0 = nb * N_BLK;
    __syncthreads();
    // b128 fill: rows contiguous in d, ROWSTRIDE is a 16B multiple
    for (int it = 0; it < (N_BLK * DIMS / 4) / BLOCK; ++it) {
      int s4 = it * BLOCK + t;
      int j = s4 >> 6, d4 = (s4 & 63) * 4;
      v4f g = *(const v4f*)(w + (long long)(n0 + j) * DIMS + d4);
      *(v4f*)(wsh + j * ROWSTRIDE + d4) = g;
    }
    __syncthreads();

#pragma unroll
    for (int nt = 0; nt < N_BLK / 16; ++nt) {
      const int nc = nt * 16;
      v8f acc = {};
      const float* ap = xs  + (mr + laneLo) * ROWSTRIDE + 2 * hi;
      const float* bp = wsh + (nc + laneLo) * ROWSTRIDE + 2 * hi;
#pragma unroll 8
      for (int k = 0; k < DIMS; k += 4) {
        v2f a  = *(const v2f*)(ap + k);   // A 16x4 frag: row=laneLo, K={2hi,2hi+1}
        v2f bb = *(const v2f*)(bp + k);   // B 4x16 frag: col=laneLo, K={2hi,2hi+1}
        acc = __builtin_amdgcn_wmma_f32_16x16x4_f32(
            false, a, false, bb, (short)0, acc, false, false);
      }
      const int ncol = n0 + nc + laneLo;
      const float wnn = wn[ncol];
#pragma unroll
      for (int i = 0; i < 8; ++i) {
        float dv = fmaf(-2.f, acc[i], xnr[i] + wnn);
        long long row = rowBase + mr + i + 8 * hi;
        __builtin_nontemporal_store(dv, &dist[row * 1024LL + ncol]);
        // branchless per-lane running min (strict < keeps first index on ties)
        bool take = dv < bestv[i];
        bestv[i] = take ? dv   : bestv[i];
        besti[i] = take ? ncol : besti[i];
      }
    }
  }

  // ---- ONE cross-lane argmin reduction at the end (16 lanes per row) ----
#pragma unroll
  for (int i = 0; i < 8; ++i) {
    float v = bestv[i]; int idx = besti[i];
#pragma unroll
    for (int m = 8; m >= 1; m >>= 1) {
      float ov = __shfl_xor(v, m, 32);
      int   oi = __shfl_xor(idx, m, 32);
      bool take = (ov < v) || (ov == v && oi < idx);  // first-index tie-break
      v   = take ? ov : v;
      idx = take ? oi : idx;
    }
    bestv[i] = v; besti[i] = idx;
  }

  // ---- indices + histogram ----
  unsigned int* counts = (unsigned int*)(wsbuf + K_CODES);
  if (laneLo == 0) {
#pragma unroll
    for (int i = 0; i < 8; ++i) {
      int m = mr + i + 8 * hi;
      rowIdx[m] = besti[i];
      encIdx[rowBase + m] = (float)besti[i];
      atomicAdd(&counts[besti[i]], 1u);
    }
  }
  __syncthreads();

  // ---- quantized (straight-through: x + (q - x)) + loss partial ----
  float lsum = 0.f;
  for (int it = 0; it < (M_TILE * DIMS) / BLOCK; ++it) {
    int e = it * BLOCK + t;
    int d = e >> 7, r = e & 127;
    float xv = xs[r * ROWSTRIDE + d];
    float qv = w[(long long)rowIdx[r] * DIMS + d];
    __builtin_nontemporal_store(xv + (qv - xv),
        &quant[((long long)b * DIMS + d) * 1024 + hw0 + r]);
    float diff = qv - xv;
    lsum = fmaf(diff, diff, lsum);
  }
#pragma unroll
  for (int m = 16; m >= 1; m >>= 1) lsum += __shfl_xor(lsum, m, 32);
  if (lane == 0) red[wave] = lsum;
  __syncthreads();
  if (t == 0) {
    float s = 0.f;
    for (int i = 0; i < 8; ++i) s += red[i];     // deterministic order
    wsbuf[2048 + blockIdx.x] = s;
  }

  // ---- one-hot encodings: region base is ==1 (mod 4) floats, so emit a
  //      3-element head, 32767 aligned float4 NT stores, 1-element tail ----
  {
    float* rbase = enc + rowBase * 1024LL;       // 131072 floats per block
    if (t < 3) rbase[t] = (t == rowIdx[0]) ? 1.f : 0.f;          // head rel 0..2
    if (t == 3) rbase[131071] = (1023 == rowIdx[127]) ? 1.f : 0.f; // tail
    for (int it = 0; it < 128; ++it) {
      int q = it * BLOCK + t;
      if (q < 32767) {
        int rel0 = 3 + 4 * q;                    // (base+rel0) is 16B aligned
        int r0 = rel0 >> 10, c0 = rel0 & 1023;
        v4f v;
#pragma unroll
        for (int j = 0; j < 4; ++j) {
          int cc = c0 + j;
          int rr = r0 + (cc >> 10);              // crosses row only at c0==1023
          cc &= 1023;
          v[j] = (cc == rowIdx[rr]) ? 1.f : 0.f;
        }
        __builtin_nontemporal_store(v, (v4f*)(rbase + rel0));
      }
    }
  }
}

__global__ __launch_bounds__(256) void vq_finalize(const float* __restrict__ wsbuf,
                                                   float* __restrict__ out) {
  __shared__ float red[256];
  int t = threadIdx.x;
  const unsigned int* counts = (const unsigned int*)(wsbuf + K_CODES);
  float h = 0.f;
  for (int k = t; k < K_CODES; k += 256) {
    float p = (float)counts[k] * (1.f / 65536.f);
    h += p * logf(p + 1e-10f);
  }
  red[t] = h;
  __syncthreads();
  if (t == 0) {
    float H = 0.f;
    for (int i = 0; i < 256; ++i) H += red[i];
    out[151060481LL] = expf(-H);                 // perplexity
    float s = 0.f;
    for (int i = 0; i < 512; ++i) s += wsbuf[2048 + i];
    float mse = s * (1.f / 16777216.f);          // mean over 64*256*1024
    out[83886080LL] = 1.25f * mse;               // q_loss + 0.25*e_loss
  }
}

extern "C" void kernel_launch(void* const* d_in, const int* in_sizes, int n_in,
                              void* d_out, int out_size, void* d_ws, size_t ws_size,
                              hipStream_t stream) {
  const float* x = (const float*)d_in[0];
  const float* w = (const float*)d_in[1];
  float* out = (float*)d_out;
  float* wsb = (float*)d_ws;

  vq_prep<<<4, 256, 0, stream>>>(w, wsb);
  size_t ldsBytes =
      (size_t)(M_TILE * ROWSTRIDE + N_BLK * ROWSTRIDE + M_TILE + K_CODES + M_TILE + 8) *
      sizeof(float);                              // ~205 KB of the 320 KB WGP LDS
  vq_main<<<N_ROWS / M_TILE, BLOCK, ldsBytes, stream>>>(x, w, out, wsb);
  vq_finalize<<<1, 256, 0, stream>>>(wsb, out);
}